// RMSWindowLayer_44461501448485
// MI455X (gfx1250) — compile-verified
//
#include <hip/hip_runtime.h>
#include <cstdint>

#define T_LEN    8192
#define WINDOW   50
#define HALO     64              // multiple of 32, >= WINDOW
#define NTHREADS 256
#define PER_THREAD (T_LEN / NTHREADS)   // 32 contiguous outputs per thread

// Padded LDS index: +1 float of pad per 32 floats.
// Lane-to-lane stride in the compute phase is 32 floats -> effective 33 after
// padding -> 33*l mod 64 is injective over 32 lanes: conflict-free LDS reads.
constexpr int padidx(int u) { return u + (u >> 5); }

__global__ __launch_bounds__(NTHREADS, 2)
void rms_window_kernel(const float* __restrict__ x, float* __restrict__ out)
{
    __shared__ float s_in[padidx(T_LEN + HALO - 1) + 1];  // ~34 KB

    const int tid = threadIdx.x;
    const long long base = (long long)blockIdx.x * T_LEN;
    const float* __restrict__ xrow = x + base;

    // ---- zero left halo (t < 0): padded slots [0, padidx(HALO-1)] ----
    if (tid <= padidx(HALO - 1)) s_in[tid] = 0.0f;

    // ---- phase 1: async global -> LDS copy of the row ----
    // Uniform compile-time trip count (32) so loop control is scalar and the
    // unroll fires. Coalesced on the global side; NT hint: input is streamed
    // once, keep L2 capacity free for output write-combining instead.
    #pragma unroll 4
    for (int it = 0; it < PER_THREAD; ++it) {
        const int u = HALO + tid + it * NTHREADS;
        unsigned lds_off = (unsigned)(uintptr_t)(&s_in[padidx(u)]);
        const float* gp = xrow + (u - HALO);
        asm volatile("global_load_async_to_lds_b32 %0, %1, off th:TH_LOAD_NT"
                     :: "v"(lds_off), "v"(gp)
                     : "memory");
    }
    asm volatile("s_wait_asynccnt 0" ::: "memory");
    __syncthreads();

    // ---- phase 2: O(1)-per-element causal running window ----
    const int t0 = tid * PER_THREAD;

    float wsum = 0.0f;
    #pragma unroll
    for (int k = 1; k < WINDOW; ++k) {
        float v = s_in[padidx(t0 + HALO - k)];
        wsum += v * v;
    }

    float* __restrict__ orow = out + base + t0;

    #pragma unroll
    for (int i = 0; i < PER_THREAD; i += 4) {
        float4 r;
        #pragma unroll
        for (int j = 0; j < 4; ++j) {
            const int t = t0 + i + j;
            float xt = s_in[padidx(t + HALO)];
            wsum += xt * xt;                         // window now covers [t-49, t]
            float inv = __builtin_amdgcn_rsqf(wsum * (1.0f / WINDOW) + 1e-6f);
            ((float*)&r)[j] = xt * inv;
            float xo = s_in[padidx(t + HALO - (WINDOW - 1))];
            wsum -= xo * xo;                         // retire x[t-49] for next t
        }
        *reinterpret_cast<float4*>(orow + i) = r;    // global_store_b128
    }
}

extern "C" void kernel_launch(void* const* d_in, const int* in_sizes, int n_in,
                              void* d_out, int out_size, void* d_ws, size_t ws_size,
                              hipStream_t stream)
{
    const float* x = (const float*)d_in[0];
    float* out = (float*)d_out;
    const int rows = in_sizes[0] / T_LEN;            // 64*64 = 4096 rows
    rms_window_kernel<<<rows, NTHREADS, 0, stream>>>(x, out);
}